// AABlock_75479755260055
// MI455X (gfx1250) — compile-verified
//
#include <hip/hip_runtime.h>
#include <stdint.h>

// ---------------------------------------------------------------------------
// Problem constants (shapes fixed by setup_inputs)
// ---------------------------------------------------------------------------
#define Bm    4
#define Sm    4
#define HWm   256
#define Ntm   257          // HW + K special
#define Cm    768
#define NHm   12
#define DHm   64
#define NGLB  1284         // HW + S*Nt tokens per batch in global attention
#define HIDm  3072

typedef __attribute__((ext_vector_type(2))) float v2f;
typedef __attribute__((ext_vector_type(8))) float v8f;

__device__ __forceinline__ float gelu_exact(float x) {
    return 0.5f * x * (1.0f + erff(x * 0.70710678118654752440f));
}

// ---------------------------------------------------------------------------
// WMMA fp32 GEMM:  Out[M,N] = A[M,K] @ W[K,N] + bias[N]  (optional exact GELU)
//
// 128x64 tile / block of 256 threads (8 waves).  Each wave owns 16 rows x 64
// cols = four 16x16 f32 accumulators, so one A fragment feeds 4 WMMAs.
// Tiles are staged into LDS with GLOBAL_LOAD_ASYNC_TO_LDS_B128 (ASYNCcnt),
// the CDNA5 direct memory->LDS path, synchronized with s_wait_asynccnt.
//
// Guard-free staging is safe by construction:
//   * Kd is always a multiple of 32 and N a multiple of 64 (768/2304/3072),
//     so only the M dimension has a tail.
//   * M-tail rows of A only influence M-tail rows of Out, which the store
//     guard discards; all A operands are followed by further live workspace,
//     so tail reads cannot fault.
// ---------------------------------------------------------------------------
#define GTM 128
#define GTN 64
#define GTK 32
#define APAD 4              // A row stride 36 floats = 144 B (16B aligned)
#define BPAD 4              // B row stride 68 floats = 272 B (16B aligned)

__device__ __forceinline__ void async_b128_to_lds(const float* gptr, void* lptr) {
    asm volatile("global_load_async_to_lds_b128 %0, %1, off"
                 :: "v"((uint32_t)(uintptr_t)lptr), "v"(gptr)
                 : "memory");
}
__device__ __forceinline__ void wait_async0() {
    asm volatile("s_wait_asynccnt 0" ::: "memory");
}

__global__ __launch_bounds__(256) void gemm_bias_act(
    const float* __restrict__ A, const float* __restrict__ W,
    const float* __restrict__ bias, float* __restrict__ Out,
    int M, int N, int Kd, int act)
{
    __shared__ __align__(16) float As[GTM][GTK + APAD];
    __shared__ __align__(16) float Bs[GTK][GTN + BPAD];

    const int tid   = threadIdx.x;
    const int wave  = tid >> 5;           // 0..7: row group of 16
    const int lane  = tid & 31;
    const int laneM = lane & 15;          // M index (A) / N index (B,C,D)
    const int laneK = (lane >> 4) << 1;   // K sub-offset per half-wave: 0 or 2
    const int tm = blockIdx.y * GTM;
    const int tn = blockIdx.x * GTN;

    v8f acc[4];
#pragma unroll
    for (int t = 0; t < 4; ++t) acc[t] = (v8f){};

    for (int k0 = 0; k0 < Kd; k0 += GTK) {
        // ---- async stage A tile 128x32 (1024 x b128) and B tile 32x64 (512)
#pragma unroll
        for (int i = 0; i < 4; ++i) {
            int idx = tid + i * 256;      // 0..1023
            int r = idx >> 3;
            int c = (idx & 7) << 2;
            async_b128_to_lds(A + (size_t)(tm + r) * Kd + (k0 + c), &As[r][c]);
        }
#pragma unroll
        for (int i = 0; i < 2; ++i) {
            int idx = tid + i * 256;      // 0..511
            int r = idx >> 4;
            int c = (idx & 15) << 2;
            async_b128_to_lds(W + (size_t)(k0 + r) * N + (tn + c), &Bs[r][c]);
        }
        // prefetch next K-chunk while this one computes
        if (k0 + GTK < Kd) {
            __builtin_prefetch(A + (size_t)(tm + (tid >> 1)) * Kd + (k0 + GTK) + (tid & 1) * 16, 0, 1);
            __builtin_prefetch(W + (size_t)(k0 + GTK + (tid >> 3)) * N + tn + (tid & 7) * 8, 0, 1);
        }
        wait_async0();
        __syncthreads();

        // ---- 8 K-steps of V_WMMA_F32_16X16X4_F32, A fragment reused 4x
#pragma unroll
        for (int kk = 0; kk < GTK; kk += 4) {
            v2f a;
            a.x = As[wave * 16 + laneM][kk + laneK];
            a.y = As[wave * 16 + laneM][kk + laneK + 1];
#pragma unroll
            for (int t = 0; t < 4; ++t) {
                v2f b;
                b.x = Bs[kk + laneK][t * 16 + laneM];
                b.y = Bs[kk + laneK + 1][t * 16 + laneM];
                acc[t] = __builtin_amdgcn_wmma_f32_16x16x4_f32(
                    false, a, false, b, (short)0, acc[t], false, false);
            }
        }
        __syncthreads();
    }

    // D layout: vgpr i -> M = i (+8 for lanes 16-31), N = lane&15
    const int mBase = tm + wave * 16 + ((lane >> 4) << 3);
#pragma unroll
    for (int t = 0; t < 4; ++t) {
        int gn = tn + t * 16 + laneM;     // N is a multiple of 64: always valid
        float bv = bias[gn];
#pragma unroll
        for (int i = 0; i < 8; ++i) {
            int gm = mBase + i;
            if (gm < M) {
                float r = acc[t][i] + bv;
                if (act) r = gelu_exact(r);
                Out[(size_t)gm * N + gn] = r;
            }
        }
    }
}

// ---------------------------------------------------------------------------
// LayerNorm with output-row remapping.
//   mode 0: out_row = (r/p0)*p1 + p2 + (r%p0)       (concat / identity)
//   mode 1: 16x16 grid -> 8x8 windows (q tokens), out rows of 64
//   mode 2: templates (257 tokens) -> 4 windows of 65, special broadcast x4
// ---------------------------------------------------------------------------
__global__ __launch_bounds__(256) void layernorm_map(
    const float* __restrict__ in, const float* __restrict__ g,
    const float* __restrict__ b, float* __restrict__ out,
    int mode, int p0, int p1, int p2)
{
    __shared__ float s1[256];
    __shared__ float s2[256];
    const int r = blockIdx.x;
    const int tid = threadIdx.x;
    const float* row = in + (size_t)r * Cm;

    float a0 = 0.f, a1 = 0.f;
    for (int c = tid; c < Cm; c += 256) { float v = row[c]; a0 += v; a1 += v * v; }
    s1[tid] = a0; s2[tid] = a1;
    __syncthreads();
    for (int s = 128; s > 0; s >>= 1) {
        if (tid < s) { s1[tid] += s1[tid + s]; s2[tid] += s2[tid + s]; }
        __syncthreads();
    }
    const float mean = s1[0] * (1.0f / Cm);
    const float var  = s2[0] * (1.0f / Cm) - mean * mean;
    const float rstd = rsqrtf(var + 1e-5f);

    int o0, o1 = -1, o2 = -1, o3 = -1;
    if (mode == 0) {
        o0 = (r / p0) * p1 + p2 + (r % p0);
    } else if (mode == 1) {
        int bb = r >> 8, n = r & 255;
        int y = n >> 4, x = n & 15;
        o0 = (bb * 4 + (y >> 3) * 2 + (x >> 3)) * 64 + (y & 7) * 8 + (x & 7);
    } else {
        int gi = r / Ntm, t = r % Ntm;
        if (t < HWm) {
            int y = t >> 4, x = t & 15;
            o0 = (gi * 4 + (y >> 3) * 2 + (x >> 3)) * 65 + (y & 7) * 8 + (x & 7);
        } else {
            o0 = (gi * 4 + 0) * 65 + 64;
            o1 = (gi * 4 + 1) * 65 + 64;
            o2 = (gi * 4 + 2) * 65 + 64;
            o3 = (gi * 4 + 3) * 65 + 64;
        }
    }
    for (int c = tid; c < Cm; c += 256) {
        float v = (row[c] - mean) * rstd * g[c] + b[c];
        out[(size_t)o0 * Cm + c] = v;
        if (o1 >= 0) {
            out[(size_t)o1 * Cm + c] = v;
            out[(size_t)o2 * Cm + c] = v;
            out[(size_t)o3 * Cm + c] = v;
        }
    }
}

// ---------------------------------------------------------------------------
// RoPE pair helper (D=32 per half, h=16)
// ---------------------------------------------------------------------------
__device__ __forceinline__ void rope_qk(
    const float* qb, const float* kb, int d, float p,
    float qv, float kv, float& qo, float& ko)
{
    int dd = d & 31;
    int j = dd & 15;
    float inv = powf(100.0f, -(float)(2 * j) * (1.0f / 32.0f));
    float ang = p * inv;
    float s, c;
    sincosf(ang, &s, &c);
    int partner = (dd < 16) ? d + 16 : d - 16;
    float qp = qb[partner], kp = kb[partner];
    if (dd < 16) { qo = qv * c - qp * s; ko = kv * c - kp * s; }
    else         { qo = qv * c + qp * s; ko = kv * c + kp * s; }
}

// Global attention q/k prep: RoPE-2D on image tokens, SO(3) on template image
// tokens (first 30 dims), pass-through elsewhere.  Output [b][h][n][d].
__global__ __launch_bounds__(64) void rope_so3_global(
    const float* __restrict__ qkv, const float* __restrict__ pos2d,
    const float* __restrict__ tpose, float* __restrict__ Qr, float* __restrict__ Kr)
{
    const int n = blockIdx.x % NGLB;
    const int b = blockIdx.x / NGLB;
    const int h = blockIdx.y;
    const int d = threadIdx.x;
    const float* base = qkv + (size_t)(b * NGLB + n) * (3 * Cm);
    const float* qb = base + h * DHm;
    const float* kb = base + Cm + h * DHm;
    float qv = qb[d], kv = kb[d];
    float qo = qv, ko = kv;
    if (n < HWm) {
        float p = pos2d[((size_t)b * HWm + n) * 2 + (d < 32 ? 0 : 1)];
        rope_qk(qb, kb, d, p, qv, kv, qo, ko);
    } else {
        int m = n - HWm;
        int s_ = m / Ntm, t = m % Ntm;
        if (t < HWm && d < 30) {
            int i = d % 3, g0 = d - i;
            const float* R = tpose + (size_t)(b * Sm + s_) * 16;  // 4x4 row-major
            float q0 = qb[g0], q1 = qb[g0 + 1], q2 = qb[g0 + 2];
            float k0 = kb[g0], k1 = kb[g0 + 1], k2 = kb[g0 + 2];
            qo = R[i * 4 + 0] * q0 + R[i * 4 + 1] * q1 + R[i * 4 + 2] * q2;
            ko = R[i * 4 + 0] * k0 + R[i * 4 + 1] * k1 + R[i * 4 + 2] * k2;
        }
    }
    size_t o = (((size_t)b * NHm + h) * NGLB + n) * DHm + d;
    Qr[o] = qo;
    Kr[o] = ko;
}

// Local attention q/k prep: RoPE-2D on window image tokens.  Output [bw][h][i][d].
__global__ __launch_bounds__(64) void rope_local(
    const float* __restrict__ qkv, const float* __restrict__ pos2d,
    float* __restrict__ Qr, float* __restrict__ Kr,
    int Ntok, int Nimg, int winPerBatch)
{
    const int row = blockIdx.x;
    const int bw = row / Ntok, i = row % Ntok;
    const int h = blockIdx.y, d = threadIdx.x;
    const float* base = qkv + (size_t)row * (3 * Cm);
    const float* qb = base + h * DHm;
    const float* kb = base + Cm + h * DHm;
    float qv = qb[d], kv = kb[d];
    float qo = qv, ko = kv;
    if (i < Nimg) {
        int w = bw & 3, wy = w >> 1, wx = w & 1;
        int b = bw / winPerBatch;
        int iy = i >> 3, ix = i & 7;
        int n = (wy * 8 + iy) * 16 + (wx * 8 + ix);
        float p = pos2d[((size_t)b * HWm + n) * 2 + (d < 32 ? 0 : 1)];
        rope_qk(qb, kb, d, p, qv, kv, qo, ko);
    }
    size_t o = (((size_t)bw * NHm + h) * Ntok + i) * DHm + d;
    Qr[o] = qo;
    Kr[o] = ko;
}

// ---------------------------------------------------------------------------
// Flash attention (online softmax).  One query row per thread, K/V tiles in
// LDS.  Q/K layout [bi][h][n][64]; V read straight out of the QKV buffer;
// output [bi*Ntok + n][h*64 + d] ready for the projection GEMM.
// ---------------------------------------------------------------------------
#define AKT 32
__global__ __launch_bounds__(64) void flash_attn(
    const float* __restrict__ Qr, const float* __restrict__ Kr,
    const float* __restrict__ Vp, float* __restrict__ Out, int Ntok)
{
    __shared__ float Ks[AKT][DHm];
    __shared__ float Vs[AKT][DHm];
    const int bh = blockIdx.x;
    const int bi = bh / NHm, h = bh % NHm;
    const int tid = threadIdx.x;
    const int qi = blockIdx.y * 64 + tid;
    const bool valid = qi < Ntok;

    const float* qbase = Qr + (((size_t)bi * NHm + h) * Ntok) * DHm;
    const float* kbase = Kr + (((size_t)bi * NHm + h) * Ntok) * DHm;
    const float* vbase = Vp + (size_t)bi * Ntok * (3 * Cm) + h * DHm;

    float q[DHm];
#pragma unroll
    for (int d = 0; d < DHm; ++d) q[d] = valid ? qbase[(size_t)qi * DHm + d] : 0.0f;
    float acc[DHm];
#pragma unroll
    for (int d = 0; d < DHm; ++d) acc[d] = 0.0f;
    float mx = -3.0e38f, l = 0.0f;

    for (int kt = 0; kt < Ntok; kt += AKT) {
        __syncthreads();
        for (int x = tid; x < AKT * DHm; x += 64) {
            int j = x >> 6, d = x & 63;
            int kn = kt + j;
            Ks[j][d] = (kn < Ntok) ? kbase[(size_t)kn * DHm + d] : 0.0f;
            Vs[j][d] = (kn < Ntok) ? vbase[(size_t)kn * (3 * Cm) + d] : 0.0f;
        }
        __syncthreads();
        if (valid) {
            int cnt = Ntok - kt; if (cnt > AKT) cnt = AKT;
            for (int j = 0; j < cnt; ++j) {
                float s = 0.0f;
#pragma unroll
                for (int d = 0; d < DHm; ++d) s += q[d] * Ks[j][d];
                s *= 0.125f;                       // DH^-0.5
                float mn   = fmaxf(mx, s);
                float corr = __expf(mx - mn);
                float pj   = __expf(s - mn);
                l = l * corr + pj;
#pragma unroll
                for (int d = 0; d < DHm; ++d) acc[d] = acc[d] * corr + pj * Vs[j][d];
                mx = mn;
            }
        }
    }
    if (valid) {
        float inv = 1.0f / l;
        float* o = Out + ((size_t)bi * Ntok + qi) * Cm + h * DHm;
#pragma unroll
        for (int d = 0; d < DHm; ++d) o[d] = acc[d] * inv;
    }
}

// ---------------------------------------------------------------------------
// Residual scatter-adds
// ---------------------------------------------------------------------------
__global__ void add_global_residual(const float* __restrict__ proj,
                                    float* __restrict__ qc, float* __restrict__ tc)
{
    size_t idx = (size_t)blockIdx.x * blockDim.x + threadIdx.x;
    const size_t total = (size_t)Bm * NGLB * Cm;
    if (idx >= total) return;
    int c = (int)(idx % Cm);
    size_t rn = idx / Cm;
    int n = (int)(rn % NGLB);
    int b = (int)(rn / NGLB);
    float v = proj[idx];
    if (n < HWm) qc[((size_t)b * HWm + n) * Cm + c] += v;
    else {
        int m = n - HWm;
        tc[((size_t)b * (Sm * Ntm) + m) * Cm + c] += v;
    }
}

__global__ void add_qwin(const float* __restrict__ proj, float* __restrict__ qc)
{
    size_t idx = (size_t)blockIdx.x * blockDim.x + threadIdx.x;
    const size_t total = (size_t)Bm * 4 * 64 * Cm;
    if (idx >= total) return;
    int c = (int)(idx % Cm);
    size_t row = idx / Cm;
    int bw = (int)(row / 64), i = (int)(row % 64);
    int b = bw >> 2, w = bw & 3;
    int n = ((w >> 1) * 8 + (i >> 3)) * 16 + ((w & 1) * 8 + (i & 7));
    qc[((size_t)b * HWm + n) * Cm + c] += proj[idx];
}

__global__ void add_twin_img(const float* __restrict__ proj, float* __restrict__ tc)
{
    size_t idx = (size_t)blockIdx.x * blockDim.x + threadIdx.x;
    const size_t total = (size_t)Bm * Sm * 4 * 65 * Cm;
    if (idx >= total) return;
    int c = (int)(idx % Cm);
    size_t row = idx / Cm;
    int bw = (int)(row / 65), i = (int)(row % 65);
    if (i == 64) return;  // special handled separately (mean)
    int g = bw >> 2, w = bw & 3;
    int t = ((w >> 1) * 8 + (i >> 3)) * 16 + ((w & 1) * 8 + (i & 7));
    tc[((size_t)g * Ntm + t) * Cm + c] += proj[idx];
}

__global__ void add_twin_special(const float* __restrict__ proj, float* __restrict__ tc)
{
    size_t idx = (size_t)blockIdx.x * blockDim.x + threadIdx.x;
    const size_t total = (size_t)Bm * Sm * Cm;
    if (idx >= total) return;
    int c = (int)(idx % Cm);
    int g = (int)(idx / Cm);
    float s = 0.0f;
    for (int w = 0; w < 4; ++w)
        s += proj[((size_t)(g * 4 + w) * 65 + 64) * Cm + c];
    tc[((size_t)g * Ntm + HWm) * Cm + c] += 0.25f * s;
}

__global__ void add_linear(const float* __restrict__ src, float* __restrict__ dst,
                           long long n)
{
    long long idx = (long long)blockIdx.x * blockDim.x + threadIdx.x;
    if (idx < n) dst[idx] += src[idx];
}

// ---------------------------------------------------------------------------
// Host orchestration
// ---------------------------------------------------------------------------
extern "C" void kernel_launch(void* const* d_in, const int* in_sizes, int n_in,
                              void* d_out, int out_size, void* d_ws, size_t ws_size,
                              hipStream_t stream)
{
    (void)in_sizes; (void)n_in; (void)out_size; (void)ws_size;

    const float* q_in = (const float*)d_in[0];
    const float* t_in = (const float*)d_in[1];
    const float* pos  = (const float*)d_in[2];
    const float* tp   = (const float*)d_in[3];
    const float* g_qkv_w  = (const float*)d_in[4];
    const float* g_qkv_b  = (const float*)d_in[5];
    const float* g_proj_w = (const float*)d_in[6];
    const float* g_proj_b = (const float*)d_in[7];
    const float* l_qkv_w  = (const float*)d_in[8];
    const float* l_qkv_b  = (const float*)d_in[9];
    const float* l_proj_w = (const float*)d_in[10];
    const float* l_proj_b = (const float*)d_in[11];
    const float* nq1_w = (const float*)d_in[12];
    const float* nq1_b = (const float*)d_in[13];
    const float* nt1_w = (const float*)d_in[14];
    const float* nt1_b = (const float*)d_in[15];
    const float* nq2_w = (const float*)d_in[16];
    const float* nq2_b = (const float*)d_in[17];
    const float* nt2_w = (const float*)d_in[18];
    const float* nt2_b = (const float*)d_in[19];
    const float* nq3_w = (const float*)d_in[20];
    const float* nq3_b = (const float*)d_in[21];
    const float* nt3_w = (const float*)d_in[22];
    const float* nt3_b = (const float*)d_in[23];
    const float* mq_fc1_w = (const float*)d_in[24];
    const float* mq_fc1_b = (const float*)d_in[25];
    const float* mq_fc2_w = (const float*)d_in[26];
    const float* mq_fc2_b = (const float*)d_in[27];
    const float* mt_fc1_w = (const float*)d_in[28];
    const float* mt_fc1_b = (const float*)d_in[29];
    const float* mt_fc2_w = (const float*)d_in[30];
    const float* mt_fc2_b = (const float*)d_in[31];

    const size_t QE = (size_t)Bm * HWm * Cm;             // 786432
    const size_t TE = (size_t)Bm * Sm * Ntm * Cm;        // 3158016

    float* q_cur = (float*)d_out;
    float* t_cur = q_cur + QE;
    hipMemcpyAsync(q_cur, q_in, QE * sizeof(float), hipMemcpyDeviceToDevice, stream);
    hipMemcpyAsync(t_cur, t_in, TE * sizeof(float), hipMemcpyDeviceToDevice, stream);

    // workspace: bufX (LN out / attn out), bufBig (QKV / MLP hidden), rotQ/rotK
    const size_t SZ_X   = (size_t)Bm * NGLB * Cm;        // 3,944,448
    const size_t SZ_BIG = (size_t)(Bm * Sm * Ntm) * HIDm;// 12,632,064 (>= all qkv)
    float* bufX   = (float*)d_ws;
    float* bufBig = bufX + SZ_X;
    float* rotQ   = bufBig + SZ_BIG;
    float* rotK   = rotQ + SZ_X;
    float* ao     = bufX;    // reuse: LN-out consumed by QKV GEMM before attn
    float* proj   = rotQ;    // reuse: rotQ consumed by attn before proj GEMM

    dim3 blk256(256), blk64(64);
    auto gemm = [&](const float* A, const float* W, const float* bias, float* O,
                    int M, int N, int K, int act) {
        dim3 g((N + GTN - 1) / GTN, (M + GTM - 1) / GTM);
        gemm_bias_act<<<g, blk256, 0, stream>>>(A, W, bias, O, M, N, K, act);
    };
    auto ew = [&](size_t n) { return dim3((unsigned)((n + 255) / 256)); };

    // ===================== global attention =====================
    layernorm_map<<<dim3(Bm * HWm), blk256, 0, stream>>>(
        q_cur, nq1_w, nq1_b, bufX, 0, HWm, NGLB, 0);
    layernorm_map<<<dim3(Bm * Sm * Ntm), blk256, 0, stream>>>(
        t_cur, nt1_w, nt1_b, bufX, 0, Sm * Ntm, NGLB, HWm);
    gemm(bufX, g_qkv_w, g_qkv_b, bufBig, Bm * NGLB, 3 * Cm, Cm, 0);
    rope_so3_global<<<dim3(Bm * NGLB, NHm), blk64, 0, stream>>>(
        bufBig, pos, tp, rotQ, rotK);
    flash_attn<<<dim3(Bm * NHm, (NGLB + 63) / 64), blk64, 0, stream>>>(
        rotQ, rotK, bufBig + 2 * Cm, ao, NGLB);
    gemm(ao, g_proj_w, g_proj_b, proj, Bm * NGLB, Cm, Cm, 0);
    add_global_residual<<<ew(SZ_X), blk256, 0, stream>>>(proj, q_cur, t_cur);

    // ===================== local attention: q =====================
    {
        const int Mq = Bm * 4 * 64;  // 1024 window-token rows
        layernorm_map<<<dim3(Bm * HWm), blk256, 0, stream>>>(
            q_cur, nq2_w, nq2_b, bufX, 1, 0, 0, 0);
        gemm(bufX, l_qkv_w, l_qkv_b, bufBig, Mq, 3 * Cm, Cm, 0);
        rope_local<<<dim3(Mq, NHm), blk64, 0, stream>>>(
            bufBig, pos, rotQ, rotK, 64, 64, 4);
        flash_attn<<<dim3(Bm * 4 * NHm, 1), blk64, 0, stream>>>(
            rotQ, rotK, bufBig + 2 * Cm, ao, 64);
        gemm(ao, l_proj_w, l_proj_b, proj, Mq, Cm, Cm, 0);
        add_qwin<<<ew((size_t)Mq * Cm), blk256, 0, stream>>>(proj, q_cur);
    }

    // ===================== local attention: t =====================
    {
        const int Mt = Bm * Sm * 4 * 65;  // 4160 rows (64 img + 1 special per window)
        layernorm_map<<<dim3(Bm * Sm * Ntm), blk256, 0, stream>>>(
            t_cur, nt2_w, nt2_b, bufX, 2, 0, 0, 0);
        gemm(bufX, l_qkv_w, l_qkv_b, bufBig, Mt, 3 * Cm, Cm, 0);
        rope_local<<<dim3(Mt, NHm), blk64, 0, stream>>>(
            bufBig, pos, rotQ, rotK, 65, 64, 16);
        flash_attn<<<dim3(Bm * Sm * 4 * NHm, 2), blk64, 0, stream>>>(
            rotQ, rotK, bufBig + 2 * Cm, ao, 65);
        gemm(ao, l_proj_w, l_proj_b, proj, Mt, Cm, Cm, 0);
        add_twin_img<<<ew((size_t)Mt * Cm), blk256, 0, stream>>>(proj, t_cur);
        add_twin_special<<<ew((size_t)Bm * Sm * Cm), blk256, 0, stream>>>(proj, t_cur);
    }

    // ===================== MLP: q =====================
    {
        const int Mq = Bm * HWm;  // 1024
        layernorm_map<<<dim3(Mq), blk256, 0, stream>>>(
            q_cur, nq3_w, nq3_b, bufX, 0, 1 << 30, 0, 0);
        gemm(bufX, mq_fc1_w, mq_fc1_b, bufBig, Mq, HIDm, Cm, 1);
        gemm(bufBig, mq_fc2_w, mq_fc2_b, proj, Mq, Cm, HIDm, 0);
        add_linear<<<ew((size_t)Mq * Cm), blk256, 0, stream>>>(
            proj, q_cur, (long long)Mq * Cm);
    }

    // ===================== MLP: t =====================
    {
        const int Mt = Bm * Sm * Ntm;  // 4112
        layernorm_map<<<dim3(Mt), blk256, 0, stream>>>(
            t_cur, nt3_w, nt3_b, bufX, 0, 1 << 30, 0, 0);
        gemm(bufX, mt_fc1_w, mt_fc1_b, bufBig, Mt, HIDm, Cm, 1);
        gemm(bufBig, mt_fc2_w, mt_fc2_b, proj, Mt, Cm, HIDm, 0);
        add_linear<<<ew((size_t)Mt * Cm), blk256, 0, stream>>>(
            proj, t_cur, (long long)Mt * Cm);
    }
}